// BoxFilter_1580547966992
// MI455X (gfx1250) — compile-verified
//
#include <hip/hip_runtime.h>

typedef __attribute__((ext_vector_type(2))) float v2f;
typedef __attribute__((ext_vector_type(8))) float v8f;

#define HH     512
#define WW     512
#define NIMG   128          // N*C = 8*16
#define RAD    8
#define TILES  32           // 512/16
#define INVK   (1.0f/17.0f)

__device__ __forceinline__ int clampi(int v, int lo, int hi) {
    return v < lo ? lo : (v > hi ? hi : v);
}

// band value: 1/17 iff kk in [m, m+16]  (single unsigned compare)
__device__ __forceinline__ float band(int kk, int m) {
    return ((unsigned)(kk - m) <= 2u * RAD) ? INVK : 0.0f;
}

// ---------------------------------------------------------------------------
// Pass 1 (horizontal): T = X * A_band.   One wave per 16x16 tile of T.
// A-operand = X[16 rows x 4 cols] (v2f/lane), B-operand = band slice (4x16).
// X loads non-temporal so the 192MB L2 stays reserved for the T intermediate.
// ---------------------------------------------------------------------------
__global__ __launch_bounds__(256) void box_h(const float* __restrict__ X,
                                             float* __restrict__ T)
{
    int wave = blockIdx.x * (blockDim.x >> 5) + (threadIdx.x >> 5);
    int lane = threadIdx.x & 31;

    int n   = wave / (TILES * TILES);
    int rem = wave - n * (TILES * TILES);
    int rt  = rem / TILES;
    int ct  = rem - rt * TILES;

    const float* img  = X + (size_t)n * (HH * WW);
    float*       timg = T + (size_t)n * (HH * WW);

    int r0 = rt * 16, c0 = ct * 16;
    int m  = lane & 15;              // A: row-in-tile, B: out col-in-tile
    int kb = (lane < 16) ? 0 : 2;    // K pair base inside each K=4 step

    const float* rowp = img + (size_t)(r0 + m) * WW;

    v8f acc = {0.f, 0.f, 0.f, 0.f, 0.f, 0.f, 0.f, 0.f};

    if (ct >= 1 && ct <= TILES - 2) {
        // ---- interior fast path: no clamps, no in-image masks, b64 loads ---
#pragma unroll
        for (int s = 0; s < 8; ++s) {
            int kk = 4 * s + kb;
            // 8-byte aligned contiguous pair -> one global_load_b64 (NT)
            v2f a = __builtin_nontemporal_load(
                        (const v2f*)(rowp + (c0 - RAD + kk)));
            v2f b;
            b.x = band(kk,     m);
            b.y = band(kk + 1, m);
            acc = __builtin_amdgcn_wmma_f32_16x16x4_f32(false, a, false, b,
                                                        (short)0, acc, false, false);
        }
    } else {
        // ---- edge path: clamp addresses, band zeroes out-of-image terms ----
#pragma unroll
        for (int s = 0; s < 8; ++s) {
            int kk   = 4 * s + kb;
            int col0 = c0 - RAD + kk;
            int col1 = col0 + 1;
            v2f a, b;
            a.x = __builtin_nontemporal_load(rowp + clampi(col0, 0, WW - 1));
            a.y = __builtin_nontemporal_load(rowp + clampi(col1, 0, WW - 1));
            b.x = (col0 >= 0 && col0 < WW) ? band(kk,     m) : 0.0f;
            b.y = (col1 >= 0 && col1 < WW) ? band(kk + 1, m) : 0.0f;
            acc = __builtin_amdgcn_wmma_f32_16x16x4_f32(false, a, false, b,
                                                        (short)0, acc, false, false);
        }
    }

    // D layout: M = v + (lane<16 ? 0 : 8), N = lane&15
    float* outp = timg + (size_t)(r0 + ((lane < 16) ? 0 : 8)) * WW + (c0 + m);
#pragma unroll
    for (int v = 0; v < 8; ++v)
        outp[(size_t)v * WW] = acc[v];
}

// ---------------------------------------------------------------------------
// Pass 2 (vertical): Out = A_band * T.   One wave per 16x16 tile of Out.
// A-operand = band slice (16x4), B-operand = T[4 rows x 16 cols].
// T reads hit L2 (written by pass 1); final stores are non-temporal streams.
// ---------------------------------------------------------------------------
__global__ __launch_bounds__(256) void box_v(const float* __restrict__ T,
                                             float* __restrict__ O)
{
    int wave = blockIdx.x * (blockDim.x >> 5) + (threadIdx.x >> 5);
    int lane = threadIdx.x & 31;

    int n   = wave / (TILES * TILES);
    int rem = wave - n * (TILES * TILES);
    int rt  = rem / TILES;
    int ct  = rem - rt * TILES;

    const float* timg = T + (size_t)n * (HH * WW);
    float*       oimg = O + (size_t)n * (HH * WW);

    int r0 = rt * 16, c0 = ct * 16;
    int m  = lane & 15;              // A: out row-in-tile, B: col-in-tile
    int kb = (lane < 16) ? 0 : 2;

    const float* colp = timg + (c0 + m);

    v8f acc = {0.f, 0.f, 0.f, 0.f, 0.f, 0.f, 0.f, 0.f};

    if (rt >= 1 && rt <= TILES - 2) {
        // ---- interior fast path ----
#pragma unroll
        for (int s = 0; s < 8; ++s) {
            int kk   = 4 * s + kb;
            int row0 = r0 - RAD + kk;
            v2f a, b;
            a.x = band(kk,     m);
            a.y = band(kk + 1, m);
            b.x = colp[(size_t)row0 * WW];
            b.y = colp[(size_t)(row0 + 1) * WW];
            acc = __builtin_amdgcn_wmma_f32_16x16x4_f32(false, a, false, b,
                                                        (short)0, acc, false, false);
        }
    } else {
        // ---- edge path ----
#pragma unroll
        for (int s = 0; s < 8; ++s) {
            int kk   = 4 * s + kb;
            int row0 = r0 - RAD + kk;
            int row1 = row0 + 1;
            v2f a, b;
            a.x = (row0 >= 0 && row0 < HH) ? band(kk,     m) : 0.0f;
            a.y = (row1 >= 0 && row1 < HH) ? band(kk + 1, m) : 0.0f;
            b.x = colp[(size_t)clampi(row0, 0, HH - 1) * WW];
            b.y = colp[(size_t)clampi(row1, 0, HH - 1) * WW];
            acc = __builtin_amdgcn_wmma_f32_16x16x4_f32(false, a, false, b,
                                                        (short)0, acc, false, false);
        }
    }

    float* outp = oimg + (size_t)(r0 + ((lane < 16) ? 0 : 8)) * WW + (c0 + m);
#pragma unroll
    for (int v = 0; v < 8; ++v)
        __builtin_nontemporal_store(acc[v], outp + (size_t)v * WW);
}

// ---------------------------------------------------------------------------
extern "C" void kernel_launch(void* const* d_in, const int* in_sizes, int n_in,
                              void* d_out, int out_size, void* d_ws, size_t ws_size,
                              hipStream_t stream)
{
    (void)in_sizes; (void)n_in; (void)out_size; (void)ws_size;

    const float* x   = (const float*)d_in[0];   // (8,16,512,512) f32; r=8 fixed
    float*       out = (float*)d_out;
    float*       tmp = (float*)d_ws;            // needs 128 MB scratch for T

    const int waves  = NIMG * TILES * TILES;    // 131072 tiles per pass
    dim3 block(256);                            // 8 waves per block
    dim3 grid(waves / 8);                       // 16384 blocks

    box_h<<<grid, block, 0, stream>>>(x, tmp);
    box_v<<<grid, block, 0, stream>>>(tmp, out);
}